// FALayer_4784593568250
// MI455X (gfx1250) — compile-verified
//
#include <hip/hip_runtime.h>
#include <math.h>

// -------- WMMA vector types (CDNA5, wave32) --------
typedef __attribute__((ext_vector_type(2))) float v2f;
typedef __attribute__((ext_vector_type(8))) float v8f;

#define D_FEAT 128

// ---------------------------------------------------------------------------
// Kernel 0: zero the output accumulator (d_out is poisoned by the harness).
// ---------------------------------------------------------------------------
__global__ void fal_zero_kernel(float4* __restrict__ p, int n4) {
    int i = blockIdx.x * blockDim.x + threadIdx.x;
    int stride = gridDim.x * blockDim.x;
    float4 zf = {0.0f, 0.0f, 0.0f, 0.0f};
    for (; i < n4; i += stride) p[i] = zf;
}

// ---------------------------------------------------------------------------
// Kernel 1: node projections via V_WMMA_F32_16X16X4_F32 (full f32 precision).
//   proj[0..N)   = h @ w_dst   (gate_w[0..128))
//   proj[N..2N)  = h @ w_src   (gate_w[128..256))
// One wave per 16-row tile of h. A = 16x4 tile of h, B = 4x16 with w_dst in
// column 0 and w_src in column 1 (other columns zero). Loop K=0..127 step 4,
// accumulating C (16x16 f32, 8 VGPRs). Per ISA 7.12.2:
//   A (16x4 f32): lanes 0-15 -> M=lane, V0=K+0, V1=K+1; lanes 16-31 -> K+2,K+3
//   B (4x16 f32): lanes 0-15 -> N=lane, V0=K+0, V1=K+1; lanes 16-31 -> K+2,K+3
//   C/D: VGPR v holds (M=v, N=lane) for lanes 0-15 and (M=v+8) for lanes 16-31
// ---------------------------------------------------------------------------
__global__ void fal_proj_wmma_kernel(const float* __restrict__ h,
                                     const float* __restrict__ gate_w,
                                     float* __restrict__ proj,
                                     int N) {
    const int lane  = threadIdx.x & 31;
    const int tile  = (int)((blockIdx.x * blockDim.x + threadIdx.x) >> 5);
    const int ntile = (N + 15) >> 4;
    if (tile >= ntile) return;   // wave-uniform: EXEC all-1 inside, all-0 skipped

    const int r0    = tile << 4;
    const int mrow  = lane & 15;            // A row within tile
    const int khalf = (lane >> 4) << 1;     // 0 for lanes 0-15, 2 for lanes 16-31
    // Clamp row for tail tiles (loads only; stores are guarded below).
    int row = r0 + mrow; if (row >= N) row = N - 1;
    const float* __restrict__ hrow = h + (size_t)row * D_FEAT;

    // B column selector: col 0 -> w_dst, col 1 -> w_src, cols 2..15 -> zero
    const int ncol = lane & 15;
    const float* __restrict__ wsel = (ncol == 0) ? gate_w : (gate_w + D_FEAT);
    const float wscale = (ncol < 2) ? 1.0f : 0.0f;

    v8f c = {0.f, 0.f, 0.f, 0.f, 0.f, 0.f, 0.f, 0.f};

    for (int kk = 0; kk < D_FEAT; kk += 4) {
        v2f a, b;
        a.x = hrow[kk + khalf];
        a.y = hrow[kk + khalf + 1];
        b.x = wsel[kk + khalf] * wscale;
        b.y = wsel[kk + khalf + 1] * wscale;
        // 8 args: (neg_a, A, neg_b, B, c_mod, C, reuse_a, reuse_b)
        c = __builtin_amdgcn_wmma_f32_16x16x4_f32(
                false, a, false, b, (short)0, c, false, false);
    }

    // Columns 0 and 1 of the 16x16 result hold proj_dst / proj_src.
    if (ncol < 2) {
        float* __restrict__ out = proj + (size_t)ncol * N;
        const int rbase = r0 + ((lane >> 4) << 3);   // +0 or +8
#pragma unroll
        for (int v = 0; v < 8; ++v) {
            int r = rbase + v;
            if (r < N) out[r] = c[v];
        }
    }
}

// ---------------------------------------------------------------------------
// Kernel 2: edge gather -> gate -> scatter-add. One wave per edge:
//   lane L handles floats [4L, 4L+4) of the 128-wide row (float4 load,
//   4x global_atomic_add_f32). Gate operands are wave-uniform addresses
//   (single L2 line broadcast). Working set is L2-resident on MI455X.
// ---------------------------------------------------------------------------
__global__ void fal_edge_kernel(const float* __restrict__ h,
                                const float* __restrict__ dnode,
                                const int* __restrict__ src,
                                const int* __restrict__ dst,
                                const float* __restrict__ proj,   // [2*N]
                                const float* __restrict__ gate_b, // [1]
                                float* __restrict__ z,
                                int E, int N) {
    const int lane   = threadIdx.x & 31;
    const int wave   = (int)((blockIdx.x * blockDim.x + threadIdx.x) >> 5);
    const int nwaves = (int)((gridDim.x * blockDim.x) >> 5);
    const float b = gate_b[0];

    for (int e = wave; e < E; e += nwaves) {
        const int s = src[e];
        const int t = dst[e];
        const float g    = tanhf(proj[t] + proj[N + s] + b);
        const float coef = g * dnode[t] * dnode[s];

        const float4* __restrict__ hs =
            (const float4*)(h + (size_t)s * D_FEAT);
        float4 v = hs[lane];                       // 512B row, fully coalesced

        float* __restrict__ zp = z + (size_t)t * D_FEAT + (lane << 2);
        atomicAdd(zp + 0, v.x * coef);
        atomicAdd(zp + 1, v.y * coef);
        atomicAdd(zp + 2, v.z * coef);
        atomicAdd(zp + 3, v.w * coef);
    }
}

// ---------------------------------------------------------------------------
// Host-side launcher (graph-capture safe: only kernel launches on `stream`).
// Input order: h (N*D), d (N), src (E), dst (E), gate_w (2D), gate_b (1).
// ---------------------------------------------------------------------------
extern "C" void kernel_launch(void* const* d_in, const int* in_sizes, int n_in,
                              void* d_out, int out_size, void* d_ws, size_t ws_size,
                              hipStream_t stream) {
    const float* h      = (const float*)d_in[0];
    const float* dnode  = (const float*)d_in[1];
    const int*   src    = (const int*)d_in[2];
    const int*   dst    = (const int*)d_in[3];
    const float* gate_w = (const float*)d_in[4];
    const float* gate_b = (const float*)d_in[5];

    const int N = in_sizes[1];
    const int E = in_sizes[2];

    float* z    = (float*)d_out;
    float* proj = (float*)d_ws;          // 2*N floats (1.6 MB) of scratch

    // 1) zero the accumulator
    const int n4 = out_size >> 2;        // N*D/4 float4s
    fal_zero_kernel<<<2048, 256, 0, stream>>>((float4*)z, n4);

    // 2) WMMA projections: one wave per 16-row tile
    const int ntile  = (N + 15) >> 4;
    const int wpb    = 256 / 32;         // 8 waves per block
    const int pblk   = (ntile + wpb - 1) / wpb;
    fal_proj_wmma_kernel<<<pblk, 256, 0, stream>>>(h, gate_w, proj, N);

    // 3) edge gather/gate/scatter: wave per edge, grid-stride over edges
    fal_edge_kernel<<<1536, 256, 0, stream>>>(h, dnode, src, dst, proj,
                                              gate_b, z, E, N);
}